// HandGraphConvNet_51745765982594
// MI455X (gfx1250) — compile-verified
//
#include <hip/hip_runtime.h>

// CDNA5 / gfx1250 fused HandGraphConvNet forward.
// out = BN2( A @ ( relu(BN1( (A@H) @ W1 + b1 )) @ W2 ) + b2 ) + x
// Aggregations run on V_WMMA_F32_16X16X4_F32; transforms (K=2/N=2) on VALU.

typedef float v2f __attribute__((ext_vector_type(2)));
typedef float v8f __attribute__((ext_vector_type(8)));

#define NJ      21      // joints
#define CJ      42      // joints * 2 channels
#define NP      48      // padded to 3x16 (N) / 12x4 (K)
#define HID     64      // hidden channels
#define ROWS    64      // (s,b) rows per workgroup (4 waves * 16)
#define THREADS 128     // 4 wave32s
#define BN_EPS  1e-5f

__launch_bounds__(THREADS)
__global__ void gcn_fused_kernel(
    const float* __restrict__ x,   const float* __restrict__ adj,
    const float* __restrict__ W1,  const float* __restrict__ b1,
    const float* __restrict__ W2,  const float* __restrict__ b2,
    const float* __restrict__ g1,  const float* __restrict__ be1,
    const float* __restrict__ m1,  const float* __restrict__ v1,
    const float* __restrict__ g2,  const float* __restrict__ be2,
    const float* __restrict__ m2,  const float* __restrict__ v2,
    float* __restrict__ out)
{
    // Aexp = A (x) I2, zero-padded to 48x48; symmetric since A is symmetric.
    __shared__ float ldsA[NP][NP];
    __shared__ float ldsX[ROWS][NP];   // input rows (residual source too)
    __shared__ float ldsT[ROWS][NP];   // T = X@Aexp, then overwritten by P' in place
    __shared__ float cW1[2][HID];      // W1 * scale1
    __shared__ float cb1[HID];         // (b1 - m1)*scale1 + be1
    __shared__ float cW2[HID][2];      // W2 * scale2  (scale2 commutes with agg)
    __shared__ float cb2[2];           // (b2 - m2)*scale2 + be2

    const int tid  = threadIdx.x;
    const int row0 = blockIdx.x * ROWS;

    // ---------- Phase 0: constants into LDS ----------
    for (int p = tid; p < NP * NP; p += THREADS) {
        const int r = p / NP, c = p % NP;
        float v = 0.0f;
        if (r < CJ && c < CJ && (((r ^ c) & 1) == 0))
            v = adj[(r >> 1) * NJ + (c >> 1)];
        ldsA[r][c] = v;
    }
    for (int d = tid; d < HID; d += THREADS) {
        const float s1 = g1[d] * rsqrtf(v1[d] + BN_EPS);
        cW1[0][d] = W1[0 * HID + d] * s1;
        cW1[1][d] = W1[1 * HID + d] * s1;
        cb1[d]    = (b1[d] - m1[d]) * s1 + be1[d];
    }
    for (int p = tid; p < HID * 2; p += THREADS) {
        const int d = p >> 1, c = p & 1;
        const float s2 = g2[c] * rsqrtf(v2[c] + BN_EPS);
        cW2[d][c] = W2[d * 2 + c] * s2;
    }
    if (tid < 2) {
        const float s2 = g2[tid] * rsqrtf(v2[tid] + BN_EPS);
        cb2[tid] = (b2[tid] - m2[tid]) * s2 + be2[tid];
    }

    // ---------- Phase 1: stage 64 rows of x ----------
    const float* xblk = x + (size_t)row0 * CJ;
    for (int p = tid; p < ROWS * NP; p += THREADS) {
        const int r = p / NP, c = p % NP;
        ldsX[r][c] = (c < CJ) ? xblk[r * CJ + c] : 0.0f;
    }
    __syncthreads();

    // WMMA lane decomposition (wave32):
    //  A frag (16x4 f32):  M = lane%16, K = 2*(lane/16) + v   (v = vgpr 0/1)
    //  B frag (4x16 f32):  N = lane%16, K = 2*(lane/16) + v
    //  C/D   (16x16 f32):  vgpr r -> M = r + 8*(lane/16), N = lane%16
    const int wave  = tid >> 5;
    const int lane  = tid & 31;
    const int lhalf = lane >> 4;
    const int l16   = lane & 15;
    const int wrow  = wave * 16;

    // ---------- Phase 2: T = X @ Aexp  (layer-1 aggregation) ----------
    v8f acc[3] = {};
    for (int k0 = 0; k0 < NP; k0 += 4) {
        const v2f afrag = *(const v2f*)&ldsX[wrow + l16][k0 + 2 * lhalf];
#pragma unroll
        for (int nt = 0; nt < 3; ++nt) {
            // Aexp symmetric: B[k][n] = Aexp[n][k] -> contiguous float2 load
            const v2f bfrag = *(const v2f*)&ldsA[nt * 16 + l16][k0 + 2 * lhalf];
            acc[nt] = __builtin_amdgcn_wmma_f32_16x16x4_f32(
                false, afrag, false, bfrag, (short)0, acc[nt], false, false);
        }
    }
#pragma unroll
    for (int nt = 0; nt < 3; ++nt)
#pragma unroll
        for (int r = 0; r < 8; ++r)
            ldsT[wrow + r + 8 * lhalf][nt * 16 + l16] = acc[nt][r];
    __syncthreads();

    // ---------- Phase 3: per-(row,joint) 2 -> 64 -> 2 transform (VALU) ----------
    // u[d] = relu(t0*cW1[0][d] + t1*cW1[1][d] + cb1[d]);  p[c] = sum_d u[d]*cW2[d][c]
    for (int p = tid; p < ROWS * NJ; p += THREADS) {
        const int r = p / NJ;
        const int i = p - r * NJ;
        const float t0 = ldsT[r][2 * i + 0];
        const float t1 = ldsT[r][2 * i + 1];
        float p0 = 0.0f, p1 = 0.0f;
#pragma unroll 8
        for (int d = 0; d < HID; ++d) {
            const float o = fmaf(t0, cW1[0][d], fmaf(t1, cW1[1][d], cb1[d]));
            const float u = fmaxf(o, 0.0f);
            p0 = fmaf(u, cW2[d][0], p0);
            p1 = fmaf(u, cW2[d][1], p1);
        }
        ldsT[r][2 * i + 0] = p0;   // in place: only this thread touches these
        ldsT[r][2 * i + 1] = p1;   // (padded cols 42..47 remain exactly 0)
    }
    __syncthreads();

    // ---------- Phase 4: O = P' @ Aexp  (layer-2 aggregation) + epilogue ----------
    v8f acc2[3] = {};
    for (int k0 = 0; k0 < NP; k0 += 4) {
        const v2f afrag = *(const v2f*)&ldsT[wrow + l16][k0 + 2 * lhalf];
#pragma unroll
        for (int nt = 0; nt < 3; ++nt) {
            const v2f bfrag = *(const v2f*)&ldsA[nt * 16 + l16][k0 + 2 * lhalf];
            acc2[nt] = __builtin_amdgcn_wmma_f32_16x16x4_f32(
                false, afrag, false, bfrag, (short)0, acc2[nt], false, false);
        }
    }
#pragma unroll
    for (int nt = 0; nt < 3; ++nt) {
        const int col = nt * 16 + l16;
        if (col < CJ) {
            const float cc = cb2[col & 1];
#pragma unroll
            for (int r = 0; r < 8; ++r) {
                const int m = r + 8 * lhalf;
                out[(size_t)(row0 + wrow + m) * CJ + col] =
                    acc2[nt][r] + cc + ldsX[wrow + m][col];
            }
        }
    }
}

extern "C" void kernel_launch(void* const* d_in, const int* in_sizes, int n_in,
                              void* d_out, int out_size, void* d_ws, size_t ws_size,
                              hipStream_t stream) {
    const float* x   = (const float*)d_in[0];
    const float* adj = (const float*)d_in[1];
    const float* W1  = (const float*)d_in[2];
    const float* b1  = (const float*)d_in[3];
    const float* W2  = (const float*)d_in[4];
    const float* b2  = (const float*)d_in[5];
    const float* g1  = (const float*)d_in[6];
    const float* be1 = (const float*)d_in[7];
    const float* m1  = (const float*)d_in[8];
    const float* v1  = (const float*)d_in[9];
    const float* g2  = (const float*)d_in[10];
    const float* be2 = (const float*)d_in[11];
    const float* m2  = (const float*)d_in[12];
    const float* v2  = (const float*)d_in[13];
    float* out = (float*)d_out;

    const int rows   = in_sizes[0] / CJ;   // S*B = 32768
    const int blocks = rows / ROWS;        // 512
    gcn_fused_kernel<<<blocks, THREADS, 0, stream>>>(
        x, adj, W1, b1, W2, b2, g1, be1, m1, v1, g2, be2, m2, v2, out);
}